// GIN_attn_20641612824580
// MI455X (gfx1250) — compile-verified
//
#include <hip/hip_runtime.h>

#define DIM    74
#define DPAD   80
#define TILE_M 16

typedef __attribute__((ext_vector_type(2))) float v2f;
typedef __attribute__((ext_vector_type(8))) float v8f;

// ---------------- Kernel 1: zero the aggregation buffer ----------------
__global__ __launch_bounds__(256) void gin_zero_kernel(float* __restrict__ p, int n) {
  int i = blockIdx.x * blockDim.x + threadIdx.x;
  int stride = gridDim.x * blockDim.x;
  for (; i < n; i += stride) p[i] = 0.0f;
}

// ---------------- Kernel 2: edge gather + atomic scatter-add ----------------
// Each wave grabs 32 consecutive edges with ONE coalesced load of src/dst per
// lane, then broadcasts each (s,d) pair across the wave via __shfl (lane
// permute) and sweeps the 74 feature dims in 3 coalesced 128B passes.
// feats/agg rows are L2-resident (29.6 MB each vs 192 MB L2); only the edge
// lists stream from HBM.
__global__ __launch_bounds__(256) void gin_edge_kernel(const float* __restrict__ feats,
                                                       const int*   __restrict__ src,
                                                       const int*   __restrict__ dst,
                                                       float*       __restrict__ agg,
                                                       int nEdges) {
  const int lane    = threadIdx.x & 31;
  const int wave    = (blockIdx.x * blockDim.x + threadIdx.x) >> 5;
  const int nWaves  = (gridDim.x * blockDim.x) >> 5;
  const int nGroups = (nEdges + 31) >> 5;

  for (int g = wave; g < nGroups; g += nWaves) {
    const int eBase = g << 5;
    const int eLane = eBase + lane;
    // one coalesced 128B load per list per 32 edges
    const int myS = (eLane < nEdges) ? src[eLane] : 0;
    const int myD = (eLane < nEdges) ? dst[eLane] : 0;

    const int count = min(32, nEdges - eBase);
    for (int i = 0; i < count; ++i) {
      const int s = __shfl(myS, i, 32);
      const int d = __shfl(myD, i, 32);
      const float* __restrict__ frow = feats + (size_t)s * DIM;
      float*       __restrict__ arow = agg   + (size_t)d * DIM;
#pragma unroll
      for (int base = 0; base < DIM; base += 32) {
        const int k = base + lane;
        if (k < DIM) {
          // hardware global_atomic_add_f32, non-returning, device scope
          __hip_atomic_fetch_add(&arow[k], frow[k], __ATOMIC_RELAXED,
                                 __HIP_MEMORY_SCOPE_AGENT);
        }
      }
    }
  }
}

// ---------------- Kernel 3: h = feats + agg ; out = elu(h @ W + b) ----------------
// Block = 160 threads = 5 waves, covers a 16-node tile.
// Wave w computes output N-tile [w*16, w*16+16) with V_WMMA_F32_16X16X4_F32,
// accumulating over 20 K-steps of the zero-padded 80-wide reduction dim.
__global__ __launch_bounds__(160) void gin_gemm_kernel(const float* __restrict__ feats,
                                                       const float* __restrict__ agg,
                                                       const float* __restrict__ W,
                                                       const float* __restrict__ bias,
                                                       float*       __restrict__ out,
                                                       int nNodes) {
  __shared__ float Ws[DPAD][DPAD + 1];   // W zero-padded to 80x80 (+1 anti-bank-conflict)
  __shared__ float Hs[TILE_M][DPAD + 1]; // (feats + agg) tile, zero-padded

  const int tid   = threadIdx.x;
  const int node0 = blockIdx.x * TILE_M;

  // Stage padded W into LDS
  for (int idx = tid; idx < DPAD * DPAD; idx += blockDim.x) {
    const int r = idx / DPAD, c = idx % DPAD;
    Ws[r][c] = (r < DIM && c < DIM) ? W[r * DIM + c] : 0.0f;
  }
  // Stage H = feats + agg tile into LDS
  for (int idx = tid; idx < TILE_M * DPAD; idx += blockDim.x) {
    const int m = idx / DPAD, c = idx % DPAD;
    const int node = node0 + m;
    float v = 0.0f;
    if (node < nNodes && c < DIM) {
      const size_t off = (size_t)node * DIM + c;
      v = feats[off] + agg[off];
    }
    Hs[m][c] = v;
  }
  __syncthreads();

  // WMMA fragment coordinates (ISA 7.12.2 layouts, wave32)
  const int wave = tid >> 5;   // 0..4 -> N-tile
  const int lane = tid & 31;
  const int half = lane >> 4;  // lane half selects K+2 (A/B) and M+8 (C/D)
  const int mn   = lane & 15;  // M for A-frag, N for B/C/D-frag
  const int n0   = wave * 16;

  v8f acc = {};
#pragma unroll
  for (int kt = 0; kt < DPAD / 4; ++kt) {
    const int k0 = kt * 4 + 2 * half;
    v2f a, bf;
    a.x  = Hs[mn][k0];          // A 16x4: lanes 0-15 hold K=k0,k0+1; 16-31 hold K=k0+2,k0+3
    a.y  = Hs[mn][k0 + 1];
    bf.x = Ws[k0][n0 + mn];     // B 4x16: row striped across lanes, half selects K+2
    bf.y = Ws[k0 + 1][n0 + mn];
    acc = __builtin_amdgcn_wmma_f32_16x16x4_f32(
        /*neg_a=*/false, a, /*neg_b=*/false, bf,
        /*c_mod=*/(short)0, acc, /*reuse_a=*/false, /*reuse_b=*/false);
  }

  const int n  = n0 + mn;
  const float bv = (n < DIM) ? bias[n] : 0.0f;
#pragma unroll
  for (int r = 0; r < 8; ++r) {
    const int m    = r + 8 * half;   // C/D: VGPR r -> M=r (lanes 0-15), M=r+8 (lanes 16-31)
    const int node = node0 + m;
    if (node < nNodes && n < DIM) {
      float v = acc[r] + bv;
      v = (v > 0.0f) ? v : (__expf(v) - 1.0f);  // ELU, alpha=1
      out[(size_t)node * DIM + n] = v;
    }
  }
}

// ---------------- Launcher ----------------
extern "C" void kernel_launch(void* const* d_in, const int* in_sizes, int n_in,
                              void* d_out, int out_size, void* d_ws, size_t ws_size,
                              hipStream_t stream) {
  const float* feats = (const float*)d_in[0];
  const int*   src   = (const int*)  d_in[1];
  const int*   dst   = (const int*)  d_in[2];
  const float* W     = (const float*)d_in[3];
  const float* bias  = (const float*)d_in[4];
  float*       out   = (float*)d_out;

  const int nNodes = in_sizes[0] / DIM;
  const int nEdges = in_sizes[1];
  float* agg = (float*)d_ws;  // nNodes * DIM floats (~29.6 MB)

  const int total = nNodes * DIM;
  gin_zero_kernel<<<2048, 256, 0, stream>>>(agg, total);

  gin_edge_kernel<<<8192, 256, 0, stream>>>(feats, src, dst, agg, nEdges);

  const int nTiles = (nNodes + TILE_M - 1) / TILE_M;
  gin_gemm_kernel<<<nTiles, 160, 0, stream>>>(feats, agg, W, bias, out, nNodes);
}